// ProposalLayerSoft_36051955482860
// MI455X (gfx1250) — compile-verified
//
#include <hip/hip_runtime.h>
#include <stdint.h>

// Volume geometry (fixed by the reference config)
#define GX 128
#define GY 128
#define GZ 64
// Tile over (x,y); z (contiguous, 64 floats = 256B) stays whole per row.
#define TX 16
#define TY 8
#define HXD (TX + 2)              // 18 halo rows in x
#define HYD (TY + 2)              // 10 halo rows in y
#define TILES_X (GX / TX)         // 8
#define TILES_Y (GY / TY)         // 16
#define NTILES (TILES_X * TILES_Y)// 128 tiles per batch
#define KCAND 10
#define NCAND (NTILES * KCAND)    // 1280 merge candidates per batch
#define THREADS 256
#define NWAVES (THREADS / 32)     // 8 wave32s per block

// ---- CDNA5 async global->LDS copy (ASYNCcnt path, no VGPR staging) ----
// Per flat-aperture rule (ISA 10.2): low 32 bits of a generic LDS pointer
// are the LDS byte address, which is exactly what VDST wants.
__device__ __forceinline__ void async_copy_b128(const float* g, float* l) {
  unsigned lds_addr = (unsigned)(size_t)l;
  unsigned long long gaddr = (unsigned long long)(size_t)g;
  asm volatile("global_load_async_to_lds_b128 %0, %1, off"
               :: "v"(lds_addr), "v"(gaddr) : "memory");
}
__device__ __forceinline__ void wait_asynccnt0() {
  asm volatile("s_wait_asynccnt 0" ::: "memory");
}

// Block-wide max of u64 keys: wave32 shuffle tree (no barriers) + 8-entry
// cross-wave LDS merge. All threads return the same value. Caller must
// barrier before the next round reuses `swave`.
__device__ __forceinline__ unsigned long long
block_max_u64(unsigned long long v, unsigned long long* swave, int tid) {
#pragma unroll
  for (int off = 16; off > 0; off >>= 1) {
    const unsigned long long o = __shfl_xor(v, off, 32);
    if (o > v) v = o;
  }
  if ((tid & 31) == 0) swave[tid >> 5] = v;
  __syncthreads();
  unsigned long long m = swave[0];
#pragma unroll
  for (int w = 1; w < NWAVES; ++w) {
    const unsigned long long o = swave[w];
    if (o > m) m = o;
  }
  return m;
}

// One 16-voxel z-chunk with compile-time ZC in {0,16,32,48}.
// Loads each of the 9 neighbor rows as aligned float4s covering the 18-value
// window; z-edges handled by constant-folded clamping (duplicating an
// in-window element leaves the 3-max identical to -inf padding).
template <int ZC>
__device__ __forceinline__ void process_chunk(const float* const* rows,
                                              unsigned rowIdx,
                                              unsigned long long (&top)[KCAND]) {
  constexpr int LO4 = (ZC == 0) ? 0 : ZC - 4;                 // 16B-aligned base
  constexpr int NLD = (ZC == 0 || ZC == 48) ? 5 : 6;          // float4 loads/row
  const float NEG_INF = __int_as_float(0xff800000);

  float v[18];
#pragma unroll
  for (int i = 0; i < 18; ++i) v[i] = NEG_INF;
  float c[16];

#pragma unroll
  for (int rsel = 0; rsel < 9; ++rsel) {
    const float4* p4 = (const float4*)(rows[rsel] + LO4);
    float a[NLD * 4];
#pragma unroll
    for (int q = 0; q < NLD; ++q) {
      const float4 f = p4[q];
      a[4 * q + 0] = f.x; a[4 * q + 1] = f.y;
      a[4 * q + 2] = f.z; a[4 * q + 3] = f.w;
    }
#pragma unroll
    for (int i = 0; i < 18; ++i) {
      int z = ZC - 1 + i;                 // constant-folded per (ZC, i)
      if (z < 0) z = 0;
      if (z > 63) z = 63;
      v[i] = fmaxf(v[i], a[z - LO4]);
    }
    if (rsel == 4) {                      // center row: keep x values
#pragma unroll
      for (int i = 0; i < 16; ++i) c[i] = a[ZC + i - LO4];
    }
  }

#pragma unroll
  for (int i = 0; i < 16; ++i) {
    const float x = c[i];
    const float m = fmaxf(fmaxf(v[i], v[i + 1]), v[i + 2]);
    const float pk = (x == m) ? x : 0.0f;
    const unsigned idx = rowIdx + (unsigned)(ZC + i);
    // key: value-major, then smaller index wins (matches lax.top_k ties)
    const unsigned long long key =
        ((unsigned long long)__float_as_uint(pk) << 32) | (0xFFFFFFFFu - idx);
    if (key > top[KCAND - 1]) {
      top[KCAND - 1] = key;
#pragma unroll
      for (int j = KCAND - 1; j > 0; --j) {
        if (top[j] > top[j - 1]) {
          unsigned long long t = top[j - 1]; top[j - 1] = top[j]; top[j] = t;
        }
      }
    }
  }
}

// Phase 1: peak mask + per-tile top-10 -> candidate keys in workspace.
__global__ void __launch_bounds__(THREADS)
proposal_phase1(const float* __restrict__ in, unsigned long long* __restrict__ cand) {
  __shared__ __align__(16) float tile[HXD * HYD * GZ];   // 46,080 B
  __shared__ unsigned long long swave[NWAVES];           //     64 B

  const int tid    = threadIdx.x;
  const int b      = blockIdx.y;
  const int tileId = blockIdx.x;
  const int tx0 = (tileId / TILES_Y) * TX;
  const int ty0 = (tileId % TILES_Y) * TY;
  const float NEG_INF = __int_as_float(0xff800000);

  // ---- stage (x,y)-halo tile into LDS: 16 lanes move one 256B z-row ----
  const int lane16 = tid & 15;
  const int mover  = tid >> 4;                 // 16 row-movers per block
  for (int row = mover; row < HXD * HYD; row += (THREADS / 16)) {
    const int hx = row / HYD;
    const int hy = row - hx * HYD;
    const int gx = tx0 - 1 + hx;
    const int gy = ty0 - 1 + hy;
    float* lrow = &tile[row * GZ];
    if ((unsigned)gx < (unsigned)GX && (unsigned)gy < (unsigned)GY) {
      const float* grow = in + ((size_t)b * GX * GY + (size_t)gx * GY + (size_t)gy) * GZ;
      async_copy_b128(grow + lane16 * 4, lrow + lane16 * 4);   // 16B per lane
    } else {
      float4 f; f.x = f.y = f.z = f.w = NEG_INF;               // -inf pad (x/y border)
      *(float4*)(lrow + lane16 * 4) = f;
    }
  }
  wait_asynccnt0();
  __syncthreads();

  // ---- per-thread 3x3x3 peak detect over 32 voxels + top-10 keys ----
  unsigned long long top[KCAND];
#pragma unroll
  for (int i = 0; i < KCAND; ++i) top[i] = 0ULL;

  // Wave-uniform z-half: waves 0-3 take z in [0,32), waves 4-7 take [32,64).
  const int r     = tid & 127;           // interior (x,y) row 0..127
  const int zhalf = tid >> 7;            // 0 or 1, uniform per wave
  const int lx = r / TY;
  const int ly = r - lx * TY;
  const int gx = tx0 + lx;
  const int gy = ty0 + ly;
  const int hx = lx + 1;
  const int hy = ly + 1;
  const unsigned rowIdx = ((unsigned)gx * GY + (unsigned)gy) * GZ;  // C-order flat base

  const float* rows[9];
#pragma unroll
  for (int dx = -1; dx <= 1; ++dx)
#pragma unroll
    for (int dy = -1; dy <= 1; ++dy)
      rows[(dx + 1) * 3 + (dy + 1)] = &tile[((hx + dx) * HYD + (hy + dy)) * GZ];

  if (zhalf == 0) {
    process_chunk<0>(rows, rowIdx, top);
    process_chunk<16>(rows, rowIdx, top);
  } else {
    process_chunk<32>(rows, rowIdx, top);
    process_chunk<48>(rows, rowIdx, top);
  }

  // ---- block-wide extraction of the tile's 10 best ----
  unsigned long long* outc = cand + ((size_t)b * NTILES + tileId) * KCAND;
  int p = 0;
  for (int rnd = 0; rnd < KCAND; ++rnd) {
    const unsigned long long mine = (p < KCAND) ? top[p] : 0ULL;
    const unsigned long long win = block_max_u64(mine, swave, tid);
    if (mine == win && mine != 0ULL) ++p;     // keys unique -> exactly one owner
    if (tid == 0) outc[rnd] = win;
    __syncthreads();                          // protect swave for next round
  }
}

// Phase 2: merge 1280 candidates per batch, global top-10, coordinate transform.
__global__ void __launch_bounds__(THREADS)
proposal_phase2(const unsigned long long* __restrict__ cand, float* __restrict__ out) {
  __shared__ unsigned long long pool[NCAND];    // 10,240 B
  __shared__ unsigned long long swave[NWAVES];
  const int tid = threadIdx.x;
  const int b   = blockIdx.x;

  for (int i = tid; i < NCAND; i += THREADS) pool[i] = cand[(size_t)b * NCAND + i];
  __syncthreads();

  for (int rnd = 0; rnd < KCAND; ++rnd) {
    unsigned long long mm = 0ULL; int mi = -1;
#pragma unroll
    for (int i = tid; i < NCAND; i += THREADS) {
      const unsigned long long k = pool[i];
      if (k > mm) { mm = k; mi = i; }
    }
    const unsigned long long win = block_max_u64(mm, swave, tid);
    if (mi >= 0 && mm == win && win != 0ULL) pool[mi] = 0ULL;   // retire winner
    if (tid == 0) {
      const float val    = __uint_as_float((unsigned)(win >> 32));
      const unsigned idx = 0xFFFFFFFFu - (unsigned)(win & 0xFFFFFFFFu);
      const float ix = (float)(idx >> 13);          // / (128*64)
      const float iy = (float)((idx >> 6) & 127u);
      const float iz = (float)(idx & 63u);
      float* o5 = out + ((size_t)b * KCAND + rnd) * 5;
      // get_real_loc: idx/(CUBE-1)*GRID + CENTER - GRID/2
      o5[0] = ix / 127.0f * 8000.0f + 0.0f   - 4000.0f;
      o5[1] = iy / 127.0f * 8000.0f + 0.0f   - 4000.0f;
      o5[2] = iz / 63.0f  * 2000.0f + 300.0f - 1000.0f;
      o5[3] = (val > 0.3f ? 1.0f : 0.0f) - 1.0f;
      o5[4] = val;
    }
    __syncthreads();   // pool retire + swave reuse visible before next scan
  }
}

extern "C" void kernel_launch(void* const* d_in, const int* in_sizes, int n_in,
                              void* d_out, int out_size, void* d_ws, size_t ws_size,
                              hipStream_t stream) {
  const float* in = (const float*)d_in[0];
  const int B = in_sizes[0] / (GX * GY * GZ);                 // 32
  unsigned long long* cand = (unsigned long long*)d_ws;       // B*1280*8 = 320 KB
  proposal_phase1<<<dim3(NTILES, B), THREADS, 0, stream>>>(in, cand);
  proposal_phase2<<<dim3(B), THREADS, 0, stream>>>(cand, (float*)d_out);
}